// _SingleDirectionGLA_16947940950474
// MI455X (gfx1250) — compile-verified
//
#include <hip/hip_runtime.h>
#include <hip/hip_bf16.h>
#include <math.h>
#include <stdint.h>

typedef float v2f __attribute__((ext_vector_type(2)));
typedef float v8f __attribute__((ext_vector_type(8)));

#define D_MODEL 1024
#define NHEADS  16
#define DKH     32
#define DVH     64
#define GATE_NORM 16.0f
#define GN_EPS  1e-5f
#define KSCALE  0.17677669529663687f  // 32^-0.5

// ---------------------------------------------------------------------------
// Async global -> LDS copy (CDNA5 GLOBAL_LOAD_ASYNC_TO_LDS_B128, ASYNCcnt).
// Builtin signature (from clang diagnostic): pointee is a 16-byte gcc-style
// int vector; src in AS1 (__device__), dst in AS3 (__shared__).
// Pointers built numerically: flat addr == AS1 addr; low 32 bits of a flat
// LDS pointer == LDS byte offset (aperture low bits are zero).
// ---------------------------------------------------------------------------
#if __has_builtin(__builtin_amdgcn_global_load_async_to_lds_b128)
#define HAVE_ASYNC_LDS 1
typedef int b128_t __attribute__((vector_size(16)));
typedef __attribute__((address_space(1))) b128_t gbl_b128_t;
typedef __attribute__((address_space(3))) b128_t lds_b128_t;
__device__ __forceinline__ void async_copy_b128(const void* gp, void* lp) {
  gbl_b128_t* g = reinterpret_cast<gbl_b128_t*>((uintptr_t)gp);
  lds_b128_t* l = reinterpret_cast<lds_b128_t*>((uint32_t)(uintptr_t)lp);
  __builtin_amdgcn_global_load_async_to_lds_b128(g, l, 0, 0);
}
#else
#define HAVE_ASYNC_LDS 0
__device__ __forceinline__ void async_copy_b128(const void* gp, void* lp) {
  *(float4*)lp = *(const float4*)gp;   // synchronous fallback
}
#endif

__device__ __forceinline__ void wait_async_all() {
#if HAVE_ASYNC_LDS
#if __has_builtin(__builtin_amdgcn_s_wait_asynccnt)
  __builtin_amdgcn_s_wait_asynccnt(0);
#else
  asm volatile("s_wait_asynccnt 0x0" ::: "memory");
#endif
#endif
}

// ---------------------------------------------------------------------------
// Tiled fp32 WMMA GEMM:  C[M,N] = act(scale * (A[M,K] @ W[K,N]) [+ bias])
// Block: 256 threads (8 waves). Block tile 64x64, K-step 16, double-buffered
// LDS fed by async b128 copies. Wave grid 4(M) x 2(N); each wave owns a
// 16x32 slab = two 16x16 accumulators via V_WMMA_F32_16X16X4_F32.
// act: 0 = none, 1 = silu(x + bias), 2 = exp(log_sigmoid(x)/GATE_NORM)
// ---------------------------------------------------------------------------
__global__ __launch_bounds__(256) void gemm_wmma_f32(
    const float* __restrict__ A, const float* __restrict__ W,
    const float* __restrict__ bias, float* __restrict__ C,
    int M, int N, int K, float scale, int act)
{
  __shared__ __align__(16) float As[2][64][20];  // 80B rows: 16B aligned, 16 banks
  __shared__ __align__(16) float Bs[2][16][68];  // 272B rows: 16B aligned

  const int tid  = threadIdx.x;
  const int lane = tid & 31;
  const int wave = tid >> 5;
  const int hlf  = lane >> 4;      // lane half
  const int lm   = lane & 15;
  const int wm   = (wave & 3) * 16;
  const int wn   = (wave >> 2) * 32;
  const int bm   = blockIdx.y * 64;
  const int bn   = blockIdx.x * 64;

  v8f acc0 = {0.f,0.f,0.f,0.f,0.f,0.f,0.f,0.f};
  v8f acc1 = {0.f,0.f,0.f,0.f,0.f,0.f,0.f,0.f};

  const int ar = tid >> 2;           // A tile row 0..63
  const int ac = (tid & 3) << 2;     // A tile col 0,4,8,12
  const int br = tid >> 4;           // B tile row 0..15
  const int bc = (tid & 15) << 2;    // B tile col 0..60
  const int arow = bm + ar;
  const int bcol = bn + bc;

  // Zero-fill slots the EXEC-masked async copies will skip (OOB rows/cols).
  const float4 z4 = {0.f, 0.f, 0.f, 0.f};
  if (arow >= M) { *(float4*)&As[0][ar][ac] = z4; *(float4*)&As[1][ar][ac] = z4; }
  if (bcol >= N) { *(float4*)&Bs[0][br][bc] = z4; *(float4*)&Bs[1][br][bc] = z4; }

  auto issue = [&](int buf, int k0) {
    if (arow < M) async_copy_b128(A + (size_t)arow * K + (k0 + ac), &As[buf][ar][ac]);
    if (bcol < N) async_copy_b128(W + (size_t)(k0 + br) * N + bcol, &Bs[buf][br][bc]);
  };

  auto compute = [&](int buf) {
#pragma unroll
    for (int kk = 0; kk < 16; kk += 4) {
      // A 16x4 fragment: lane holds row m=lm, k = kk + 2*hlf + {0,1}
      v2f a;
      a.x = As[buf][wm + lm][kk + 2*hlf + 0];
      a.y = As[buf][wm + lm][kk + 2*hlf + 1];
      // B 4x16 fragments: lane holds col n=lm, k = kk + 2*hlf + {0,1}
      v2f b0, b1;
      b0.x = Bs[buf][kk + 2*hlf + 0][wn + lm];
      b0.y = Bs[buf][kk + 2*hlf + 1][wn + lm];
      b1.x = Bs[buf][kk + 2*hlf + 0][wn + 16 + lm];
      b1.y = Bs[buf][kk + 2*hlf + 1][wn + 16 + lm];
      acc0 = __builtin_amdgcn_wmma_f32_16x16x4_f32(false, a, false, b0,
                                                   (short)0, acc0, false, false);
      acc1 = __builtin_amdgcn_wmma_f32_16x16x4_f32(false, a, false, b1,
                                                   (short)0, acc1, false, false);
    }
  };

  // Pipeline: copy(s+1) overlaps WMMA(s); asynccnt + barrier retire each stage.
  issue(0, 0);
  wait_async_all();
  __syncthreads();
  const int nstage = K >> 4;
  for (int s = 0; s < nstage; ++s) {
    const int cur = s & 1;
    if (s + 1 < nstage) issue(cur ^ 1, (s + 1) << 4);
    compute(cur);
    wait_async_all();
    __syncthreads();
  }

  // ---- epilogue: C/D layout: VGPR r -> M = r + 8*hlf, N = lm ----
#pragma unroll
  for (int r = 0; r < 8; ++r) {
    const int mm = bm + wm + r + 8*hlf;
    if (mm >= M) continue;
#pragma unroll
    for (int t = 0; t < 2; ++t) {
      const int nn = bn + wn + t*16 + lm;
      if (nn >= N) continue;
      float x = (t ? acc1[r] : acc0[r]) * scale;
      if (act == 1) {                       // silu(x + bias)
        x += bias[nn];
        x = x / (1.f + expf(-x));
      } else if (act == 2) {                // exp(log_sigmoid(x)/GATE_NORM)
        const float ls = fminf(x, 0.f) - log1pf(expf(-fabsf(x)));
        x = expf(ls * (1.f / GATE_NORM));
      }
      C[(size_t)mm * N + nn] = x;
    }
  }
}

// ---------------------------------------------------------------------------
// GLA recurrence: one block per (b,h). State[DK=32][DV=64] striped:
// lane = dk, wave (of 8) = dv octet; 8 state floats per thread in registers.
// out[dv] = sum_dk q[dk]*state[dk][dv] via wave32 butterfly reduction.
// o layout: [B,H,L,DV].
// ---------------------------------------------------------------------------
__global__ __launch_bounds__(256) void gla_scan(
    const float* __restrict__ q, const float* __restrict__ k,
    const float* __restrict__ g, const float* __restrict__ v,
    float* __restrict__ o, int L)
{
  const int bh   = blockIdx.x;          // 0..B*H-1
  const int b    = bh >> 4;
  const int h    = bh & 15;
  const int lane = threadIdx.x & 31;    // dk
  const int wave = threadIdx.x >> 5;    // dv block
  const int dv0  = wave * 8;

  const size_t qk_base = ((size_t)b * L) * (NHEADS * DKH) + h * DKH + lane;
  const size_t v_base  = ((size_t)b * L) * (NHEADS * DVH) + h * DVH + dv0;
  float* op = o + ((size_t)bh * L) * DVH + dv0;

  float st[8];
#pragma unroll
  for (int j = 0; j < 8; ++j) st[j] = 0.f;

  for (int l = 0; l < L; ++l) {
    const float qv = q[qk_base + (size_t)l * (NHEADS * DKH)];
    const float kv = k[qk_base + (size_t)l * (NHEADS * DKH)];
    const float gv = g[qk_base + (size_t)l * (NHEADS * DKH)];  // pre-exp'd gate
    const float4 va = *(const float4*)(v + v_base + (size_t)l * (NHEADS * DVH));
    const float4 vb = *(const float4*)(v + v_base + (size_t)l * (NHEADS * DVH) + 4);
    const float vv[8] = {va.x, va.y, va.z, va.w, vb.x, vb.y, vb.z, vb.w};

    float out[8];
#pragma unroll
    for (int j = 0; j < 8; ++j) {
      st[j] = st[j] * gv + kv * vv[j];
      float t = qv * st[j];
#pragma unroll
      for (int off = 16; off; off >>= 1) t += __shfl_xor(t, off, 32);
      out[j] = t;
    }
    if (lane == 0) {
      float4 o0 = {out[0], out[1], out[2], out[3]};
      float4 o1 = {out[4], out[5], out[6], out[7]};
      *(float4*)(op + (size_t)l * DVH + 0) = o0;
      *(float4*)(op + (size_t)l * DVH + 4) = o1;
    }
  }
}

// ---------------------------------------------------------------------------
// GroupNorm over DV (no affine) + gate multiply + head merge.
// One wave per (b,h,l) group of 64 values; og layout [B,L,H*DV].
// ---------------------------------------------------------------------------
__global__ __launch_bounds__(256) void groupnorm_gate(
    const float* __restrict__ o, const float* __restrict__ gate,
    float* __restrict__ og, int L, int ngroups)
{
  const int lane  = threadIdx.x & 31;
  const long gid  = (long)blockIdx.x * 8 + (threadIdx.x >> 5);  // (b,h,l) flat
  if (gid >= ngroups) return;

  const int l  = (int)(gid % L);
  const int bh = (int)(gid / L);
  const int b  = bh >> 4;
  const int h  = bh & 15;

  const float* op = o + (size_t)gid * DVH;
  const float x0 = op[lane];
  const float x1 = op[lane + 32];

  float s  = x0 + x1;
  float sq = x0 * x0 + x1 * x1;
#pragma unroll
  for (int off = 16; off; off >>= 1) {
    s  += __shfl_xor(s,  off, 32);
    sq += __shfl_xor(sq, off, 32);
  }
  const float mean = s * (1.f / 64.f);
  const float var  = sq * (1.f / 64.f) - mean * mean;
  const float rstd = rsqrtf(var + GN_EPS);

  const size_t base = ((size_t)b * L + l) * D_MODEL + h * DVH;
  og[base + lane]      = (x0 - mean) * rstd * gate[base + lane];
  og[base + lane + 32] = (x1 - mean) * rstd * gate[base + lane + 32];
}

// ---------------------------------------------------------------------------
// Host launcher
// ---------------------------------------------------------------------------
extern "C" void kernel_launch(void* const* d_in, const int* in_sizes, int n_in,
                              void* d_out, int out_size, void* d_ws, size_t ws_size,
                              hipStream_t stream) {
  (void)n_in; (void)out_size; (void)ws_size;
  const float* X   = (const float*)d_in[0];   // [B,L,D]
  const float* Wq  = (const float*)d_in[1];   // [D, D/2]
  const float* Wk  = (const float*)d_in[2];   // [D, D/2]
  const float* Wg1 = (const float*)d_in[3];   // [D, 16]
  const float* Wg2 = (const float*)d_in[4];   // [16, D/2]
  const float* Wv  = (const float*)d_in[5];   // [D, D]
  const float* Wg  = (const float*)d_in[6];   // [D, D]
  const float* bg  = (const float*)d_in[7];   // [D]
  const float* Wo  = (const float*)d_in[8];   // [D, D]
  float* out = (float*)d_out;

  const int Bq = 4;
  const int M  = in_sizes[0] / D_MODEL;       // B*L = 16384
  const int L  = M / Bq;                      // 4096

  // Workspace layout (floats)
  float* ws      = (float*)d_ws;
  float* q_buf   = ws;                                    // M*512
  float* k_buf   = q_buf  + (size_t)M * 512;              // M*512
  float* g_buf   = k_buf  + (size_t)M * 512;              // M*512 (exp(logsig/16))
  float* g1_buf  = g_buf  + (size_t)M * 512;              // M*16
  float* v_buf   = g1_buf + (size_t)M * 16;               // M*1024
  float* gate_buf= v_buf  + (size_t)M * 1024;             // M*1024
  float* o_buf   = gate_buf + (size_t)M * 1024;           // M*1024 ([B,H,L,DV])
  float* og_buf  = q_buf;  // reuse dead q+k region (exactly M*1024 floats)

  const dim3 blk(256);
  const int mt = M / 64;  // M tiles

  // Projections (all WMMA f32)
  gemm_wmma_f32<<<dim3(8,  mt), blk, 0, stream>>>(X,      Wq,  nullptr, q_buf,   M,  512, 1024, 1.0f,   0);
  gemm_wmma_f32<<<dim3(8,  mt), blk, 0, stream>>>(X,      Wk,  nullptr, k_buf,   M,  512, 1024, KSCALE, 0);
  gemm_wmma_f32<<<dim3(1,  mt), blk, 0, stream>>>(X,      Wg1, nullptr, g1_buf,  M,   16, 1024, 1.0f,   0);
  gemm_wmma_f32<<<dim3(8,  mt), blk, 0, stream>>>(g1_buf, Wg2, nullptr, g_buf,   M,  512,   16, 1.0f,   2);
  gemm_wmma_f32<<<dim3(16, mt), blk, 0, stream>>>(X,      Wv,  nullptr, v_buf,   M, 1024, 1024, 1.0f,   0);
  gemm_wmma_f32<<<dim3(16, mt), blk, 0, stream>>>(X,      Wg,  bg,      gate_buf,M, 1024, 1024, 1.0f,   1);

  // Sequential GLA recurrence: one block per (b,h)
  gla_scan<<<dim3(Bq * NHEADS), blk, 0, stream>>>(q_buf, k_buf, g_buf, v_buf, o_buf, L);

  // GroupNorm + gate + head merge
  const int ngroups = Bq * NHEADS * L;
  groupnorm_gate<<<dim3((ngroups + 7) / 8), blk, 0, stream>>>(o_buf, gate_buf, og_buf, L, ngroups);

  // Output projection
  gemm_wmma_f32<<<dim3(16, mt), blk, 0, stream>>>(og_buf, Wo, nullptr, out, M, 1024, 1024, 1.0f, 0);
}